// GraphSAGE_25589415150314
// MI455X (gfx1250) — compile-verified
//
#include <hip/hip_runtime.h>

typedef __attribute__((ext_vector_type(2))) float v2f;
typedef __attribute__((ext_vector_type(8))) float v8f;

constexpr int DF = 128;        // feature dim (D == H == 128)
constexpr int TILE_M = 16;     // node rows per block
constexpr int PITCH = 130;     // LDS row pitch in floats (bank-conflict padding)

// ---------------- degree: deg[dst] += 1 per edge -------------------------
__global__ void sage_degree_kernel(const int* __restrict__ dst,
                                   float* __restrict__ deg, int E) {
    int t = blockIdx.x * blockDim.x + threadIdx.x;
    if (t < E) atomicAdd(&deg[dst[t]], 1.0f);
}

// ---------------- scatter: sums[dst] += h[src] (one wave per edge) -------
__global__ void sage_scatter_kernel(const float* __restrict__ h,
                                    const int* __restrict__ src,
                                    const int* __restrict__ dst,
                                    float* __restrict__ sums, int E) {
    int gid  = blockIdx.x * blockDim.x + threadIdx.x;
    int e    = gid >> 5;          // wave per edge (wave32)
    int lane = gid & 31;          // lane covers 4 consecutive features
    if (e >= E) return;
    int s = src[e];
    int d = dst[e];
    const float4 v = *(const float4*)(h + (size_t)s * DF + lane * 4);
    float* p = sums + (size_t)d * DF + lane * 4;
    atomicAdd(p + 0, v.x);
    atomicAdd(p + 1, v.y);
    atomicAdd(p + 2, v.z);
    atomicAdd(p + 3, v.w);
}

// ---------------- fused SAGE layer: out = act(A@Ws + (sums/deg)@Wn + b) --
template <int DO_RELU>
__global__ __launch_bounds__(256)
void sage_gemm_kernel(const float* __restrict__ Aself,
                      const float* __restrict__ sums,
                      const float* __restrict__ deg,
                      const float* __restrict__ Wself,
                      const float* __restrict__ Wneigh,
                      const float* __restrict__ bias,
                      float* __restrict__ out,
                      int N) {
    __shared__ float As[TILE_M * PITCH];   // self features,   16 x 128
    __shared__ float An[TILE_M * PITCH];   // mean-aggregated, 16 x 128
    __shared__ float invd[TILE_M];         // per-row 1/max(deg,1)

    const int tid = threadIdx.x;
    const int rowBase = blockIdx.x * TILE_M;

    // one reciprocal per row instead of one divide per element
    if (tid < TILE_M) {
        int row = rowBase + tid;
        float dg = (row < N) ? deg[row] : 1.0f;
        invd[tid] = 1.0f / fmaxf(dg, 1.0f);
    }
    __syncthreads();

    // Cooperative stage of both A operands into LDS (reads of `out` rows in
    // layer 2 happen here, strictly before any store to those rows).
    for (int i = tid; i < TILE_M * DF; i += 256) {
        int r = i >> 7;            // i / 128
        int c = i & (DF - 1);
        int row = rowBase + r;
        float a = 0.0f, nn = 0.0f;
        if (row < N) {
            a = Aself[(size_t)row * DF + c];
            nn = sums[(size_t)row * DF + c] * invd[r];
        }
        As[r * PITCH + c] = a;
        An[r * PITCH + c] = nn;
    }
    __syncthreads();

    // Wave = one 16x16 N-tile of the 16x128 output block.
    const int wave  = tid >> 5;        // 0..7 -> N-tile
    const int lane  = tid & 31;
    const int nIdx  = lane & 15;       // A row (M) and B column within tile
    const int khalf = lane >> 4;       // 0: K={0,1}; 1: K={2,3} within K4 slice
    const int col   = wave * 16 + nIdx;

    const float bv = bias[col];
    v8f acc = {bv, bv, bv, bv, bv, bv, bv, bv};

    #pragma unroll 4
    for (int kk = 0; kk < DF / 4; ++kk) {
        const int k = kk * 4 + khalf * 2;
        // A operands (16x4 fp32 slice): lane holds A[M=nIdx][k], A[M=nIdx][k+1]
        v2f aS = *(const v2f*)&As[nIdx * PITCH + k];
        v2f aN = *(const v2f*)&An[nIdx * PITCH + k];
        // B operands (4x16 fp32 slice): lane holds W[k][col], W[k+1][col]
        v2f bS, bN;
        bS.x = Wself [(size_t)(k    ) * DF + col];
        bS.y = Wself [(size_t)(k + 1) * DF + col];
        bN.x = Wneigh[(size_t)(k    ) * DF + col];
        bN.y = Wneigh[(size_t)(k + 1) * DF + col];
        acc = __builtin_amdgcn_wmma_f32_16x16x4_f32(false, aS, false, bS,
                                                    (short)0, acc, false, false);
        acc = __builtin_amdgcn_wmma_f32_16x16x4_f32(false, aN, false, bN,
                                                    (short)0, acc, false, false);
    }

    // C/D layout: VGPR v -> M = v + 8*khalf, N = col.
    if (rowBase + TILE_M <= N) {
        // full tile: straight-line stores, no per-element guards
        float* po = out + (size_t)(rowBase + khalf * 8) * DF + col;
        #pragma unroll
        for (int v = 0; v < 8; ++v) {
            float val = acc[v];
            if (DO_RELU) val = fmaxf(val, 0.0f);
            po[(size_t)v * DF] = val;
        }
    } else {
        #pragma unroll
        for (int v = 0; v < 8; ++v) {
            int row = rowBase + v + khalf * 8;
            if (row < N) {
                float val = acc[v];
                if (DO_RELU) val = fmaxf(val, 0.0f);
                out[(size_t)row * DF + col] = val;
            }
        }
    }
}

extern "C" void kernel_launch(void* const* d_in, const int* in_sizes, int n_in,
                              void* d_out, int out_size, void* d_ws, size_t ws_size,
                              hipStream_t stream) {
    const float* x       = (const float*)d_in[0];
    const int*   src     = (const int*)  d_in[1];
    const int*   dst     = (const int*)  d_in[2];
    const float* Wself1  = (const float*)d_in[3];
    const float* Wneigh1 = (const float*)d_in[4];
    const float* b1      = (const float*)d_in[5];
    const float* Wself2  = (const float*)d_in[6];
    const float* Wneigh2 = (const float*)d_in[7];
    const float* b2      = (const float*)d_in[8];

    const int N = in_sizes[0] / DF;   // 50000
    const int E = in_sizes[1];        // 800000

    float* out  = (float*)d_out;      // holds h1 after layer 1, final after layer 2
    float* deg  = (float*)d_ws;                       // N floats
    float* sums = deg + N;                            // N*128 floats

    const int scatterBlocks = (int)(((long long)E * 32 + 255) / 256);
    const int gemmBlocks    = (N + TILE_M - 1) / TILE_M;

    // ---- layer 1 ----
    hipMemsetAsync(deg,  0, (size_t)N * sizeof(float), stream);
    hipMemsetAsync(sums, 0, (size_t)N * DF * sizeof(float), stream);
    sage_degree_kernel<<<(E + 255) / 256, 256, 0, stream>>>(dst, deg, E);
    sage_scatter_kernel<<<scatterBlocks, 256, 0, stream>>>(x, src, dst, sums, E);
    sage_gemm_kernel<1><<<gemmBlocks, 256, 0, stream>>>(x, sums, deg,
                                                        Wself1, Wneigh1, b1,
                                                        out, N);

    // ---- layer 2 (h1 lives in d_out; rows staged to LDS before overwrite) ----
    hipMemsetAsync(sums, 0, (size_t)N * DF * sizeof(float), stream);
    sage_scatter_kernel<<<scatterBlocks, 256, 0, stream>>>(out, src, dst, sums, E);
    sage_gemm_kernel<0><<<gemmBlocks, 256, 0, stream>>>(out, sums, deg,
                                                        Wself2, Wneigh2, b2,
                                                        out, N);
}